// LoFTR2Tensor_28132035789026
// MI455X (gfx1250) — compile-verified
//
#include <hip/hip_runtime.h>
#include <hip/hip_bf16.h>

// -----------------------------------------------------------------------------
// LoFTR jagged->padded-dense. Output-driven gather formulation:
//   out[b, p] = input[starts[b] + p]  if p < counts[b], else 0
// batch_indexes is sorted -> starts/ends via a single boundary-detect pass
// (no bincount/cumsum/atomics). Every output element written exactly once.
//
// Pure streaming op (~740 MB traffic, 0 FLOPs) -> HBM bound (~32 us at
// 23.3 TB/s). No matrix math -> WMMA inapplicable; optimize for widest VMEM:
// fast path is 5x global_load_b128 + 5x global_store_b128 per thread (160 B),
// all with non-temporal hints (working set ~4x the 192 MB L2).
// -----------------------------------------------------------------------------

typedef float v2f __attribute__((ext_vector_type(2)));
typedef float v4f __attribute__((ext_vector_type(4)));

// Zero starts[B] and ends[B] in workspace. Single block grid-stride loop
// (B lives only in device memory; avoids any capture-breaking host sync).
__global__ void loftr_init_ws(int* __restrict__ ws, const int* __restrict__ nb) {
    const int B = *nb;
    for (int i = threadIdx.x; i < 2 * B; i += blockDim.x) ws[i] = 0;
}

// Boundary detection over sorted batch_indexes:
//   starts[b] = first i with bi[i]==b ; ends[b] = last i + 1
__global__ void loftr_boundaries(const int* __restrict__ bi, int n,
                                 int* __restrict__ ws, const int* __restrict__ nb) {
    const int i = blockIdx.x * blockDim.x + threadIdx.x;
    if (i >= n) return;
    const int B = *nb;
    int* starts = ws;
    int* ends   = ws + B;
    const int b = bi[i];
    if (i == 0 || bi[i - 1] != b) starts[b] = i;
    if (i == n - 1 || bi[i + 1] != b) ends[b] = i + 1;
}

// One thread handles 4 consecutive padded slots.
//  - aligned fast path (src % 4 == 0): all-b128 NT loads/stores
//  - unaligned fast path: b64 keypoints + scalar conf
//  - slow path: batch boundary / padding / tail, per point with zero fill
__global__ void loftr_gather_fill(const float* __restrict__ kp0,
                                  const float* __restrict__ kp1,
                                  const float* __restrict__ conf,
                                  const int*  __restrict__ ws,
                                  const int*  __restrict__ nb,
                                  const int*  __restrict__ mp,
                                  float* __restrict__ out,
                                  unsigned total /* = B * max_points */) {
    const unsigned t    = blockIdx.x * blockDim.x + threadIdx.x;
    const unsigned base = t * 4u;
    if (base >= total) return;

    const int B = *nb;
    const unsigned L = (unsigned)*mp;
    const int* starts = ws;
    const int* ends   = ws + B;

    float* __restrict__ out_src = out;                          // (B, L, 2)
    float* __restrict__ out_tgt = out + (size_t)total * 2;      // (B, L, 2)
    float* __restrict__ out_scr = out + (size_t)total * 4;      // (B, L)

    const unsigned b = base / L;
    const unsigned p = base - b * L;
    const int s = starts[b];
    const unsigned cnt = (unsigned)(ends[b] - s);

    if (p + 4u <= L && p + 4u <= cnt && base + 4u <= total) {
        // ---- fast path: 4 valid points, same batch ----
        const size_t src = (size_t)(unsigned)s + p;
        v4f s01, s23, t01, t23, cc;

        if ((src & 3u) == 0) {
            // src 4-aligned: keypoints are 16B-aligned at float index 2*src,
            // conf is 16B-aligned at index src -> pure b128 loads.
            const v4f* k0q = (const v4f*)(kp0 + src * 2);
            const v4f* k1q = (const v4f*)(kp1 + src * 2);
            const v4f* cfq = (const v4f*)(conf + src);
            s01 = __builtin_nontemporal_load(&k0q[0]);
            s23 = __builtin_nontemporal_load(&k0q[1]);
            t01 = __builtin_nontemporal_load(&k1q[0]);
            t23 = __builtin_nontemporal_load(&k1q[1]);
            cc  = __builtin_nontemporal_load(cfq);
        } else {
            // valid but unaligned start: b64 keypoints, scalar conf
            const v2f* k0 = (const v2f*)kp0;
            const v2f* k1 = (const v2f*)kp1;
            v2f a0 = __builtin_nontemporal_load(&k0[src + 0]);
            v2f a1 = __builtin_nontemporal_load(&k0[src + 1]);
            v2f a2 = __builtin_nontemporal_load(&k0[src + 2]);
            v2f a3 = __builtin_nontemporal_load(&k0[src + 3]);
            v2f b0 = __builtin_nontemporal_load(&k1[src + 0]);
            v2f b1 = __builtin_nontemporal_load(&k1[src + 1]);
            v2f b2 = __builtin_nontemporal_load(&k1[src + 2]);
            v2f b3 = __builtin_nontemporal_load(&k1[src + 3]);
            s01 = (v4f){a0.x, a0.y, a1.x, a1.y};
            s23 = (v4f){a2.x, a2.y, a3.x, a3.y};
            t01 = (v4f){b0.x, b0.y, b1.x, b1.y};
            t23 = (v4f){b2.x, b2.y, b3.x, b3.y};
            cc  = (v4f){__builtin_nontemporal_load(&conf[src + 0]),
                        __builtin_nontemporal_load(&conf[src + 1]),
                        __builtin_nontemporal_load(&conf[src + 2]),
                        __builtin_nontemporal_load(&conf[src + 3])};
        }

        // base % 4 == 0 -> destinations are 16B/32B aligned: b128 stores legal.
        __builtin_nontemporal_store(s01, (v4f*)(out_src + (size_t)base * 2));
        __builtin_nontemporal_store(s23, (v4f*)(out_src + (size_t)base * 2 + 4));
        __builtin_nontemporal_store(t01, (v4f*)(out_tgt + (size_t)base * 2));
        __builtin_nontemporal_store(t23, (v4f*)(out_tgt + (size_t)base * 2 + 4));
        __builtin_nontemporal_store(cc,  (v4f*)(out_scr + base));
    } else {
        // ---- slow path: boundary / padding / tail ----
        #pragma unroll
        for (int j = 0; j < 4; ++j) {
            const unsigned g = base + (unsigned)j;
            if (g >= total) break;
            const unsigned bb = g / L;
            const unsigned pp = g - bb * L;
            const int ss = starts[bb];
            const unsigned cc2 = (unsigned)(ends[bb] - ss);
            float x0 = 0.f, y0 = 0.f, x1 = 0.f, y1 = 0.f, c = 0.f;
            if (pp < cc2) {
                const size_t src = (size_t)(unsigned)ss + pp;
                x0 = kp0[src * 2 + 0];
                y0 = kp0[src * 2 + 1];
                x1 = kp1[src * 2 + 0];
                y1 = kp1[src * 2 + 1];
                c  = conf[src];
            }
            out_src[(size_t)g * 2 + 0] = x0;
            out_src[(size_t)g * 2 + 1] = y0;
            out_tgt[(size_t)g * 2 + 0] = x1;
            out_tgt[(size_t)g * 2 + 1] = y1;
            out_scr[g] = c;
        }
    }
}

extern "C" void kernel_launch(void* const* d_in, const int* in_sizes, int n_in,
                              void* d_out, int out_size, void* d_ws, size_t ws_size,
                              hipStream_t stream) {
    const float* kp0  = (const float*)d_in[0];   // (N, 2) f32
    const float* kp1  = (const float*)d_in[1];   // (N, 2) f32
    const float* conf = (const float*)d_in[2];   // (N,)   f32
    const int*   bi   = (const int*)  d_in[3];   // (N,)   i32, sorted
    const int*   nb   = (const int*)  d_in[4];   // scalar: num_batches
    const int*   mp   = (const int*)  d_in[5];   // scalar: max_points

    const int n = in_sizes[2];                           // N (jagged points)
    const unsigned total = (unsigned)(out_size / 5);     // B * max_points
    int* ws = (int*)d_ws;                                // starts[B] | ends[B]
    float* out = (float*)d_out;

    // 1) zero boundary arrays (single block; B only known on device)
    loftr_init_ws<<<1, 1024, 0, stream>>>(ws, nb);

    // 2) boundary detection over sorted batch_indexes
    loftr_boundaries<<<(n + 255) / 256, 256, 0, stream>>>(bi, n, ws, nb);

    // 3) output-driven gather, 4 padded points per thread, all-b128 fast path
    const unsigned n4 = (total + 3u) / 4u;
    loftr_gather_fill<<<(n4 + 255u) / 256u, 256, 0, stream>>>(
        kp0, kp1, conf, ws, nb, mp, out, total);
}